// GAT_83356725280824
// MI455X (gfx1250) — compile-verified
//
#include <hip/hip_runtime.h>
#include <math.h>

typedef float v2f __attribute__((ext_vector_type(2)));
typedef float v8f __attribute__((ext_vector_type(8)));
typedef unsigned int uint4v __attribute__((ext_vector_type(4)));
typedef int int8v __attribute__((ext_vector_type(8)));
typedef int int4v __attribute__((ext_vector_type(4)));

#define NEG_SLOPE 0.2f

// ---------------------------------------------------------------------------
// init: zero accumulators, set running-max buffers to -inf-ish
// ---------------------------------------------------------------------------
__global__ __launch_bounds__(256) void k_init(float* out1, float* s1, float* m1,
                                              float* s2, float* m2, float* dout,
                                              int N) {
  int i = blockIdx.x * blockDim.x + threadIdx.x;
  if (i < N * 128) out1[i] = 0.0f;
  if (i < N * 8) { s1[i] = 0.0f; m1[i] = -1e30f; }
  if (i < N * 2) dout[i] = 0.0f;
  if (i < N)     { s2[i] = 0.0f; m2[i] = -1e30f; }
}

// ---------------------------------------------------------------------------
// GEMM1: h1[N,128] = x[N,128] @ W1[128,128], fp32 WMMA 16x16x4.
// A-tile (16 rows x 128 cols, 8 KB) staged into LDS with the Tensor Data
// Mover (one linear DMA per workgroup), then read via ds_load per k-step.
// 8 waves per block, each owns a 16-col tile; 32 WMMAs per wave.
// ---------------------------------------------------------------------------
__global__ __launch_bounds__(256) void k_gemm1(const float* __restrict__ x,
                                               const float* __restrict__ W,
                                               float* __restrict__ h, int n) {
  __shared__ float lds_a[16 * 128];

  const int lane = threadIdx.x & 31;
  const int waveN = threadIdx.x >> 5;       // 0..7 -> which 16-col tile
  const int m0 = blockIdx.x << 4;
  const int n0 = waveN << 4;
  const int l15 = lane & 15;
  const int hi  = lane >> 4;                // 0 or 1
  const int koff = hi << 1;                 // K offset 0 or 2 within group of 4

  int rows = n - m0;
  if (rows > 16) rows = 16;

  // --- TDM: DMA the A tile into LDS (wave 0 issues, all waves consume) ----
  if (threadIdx.x < 32) {
    unsigned long long ga =
        (unsigned long long)(const char*)(x + (size_t)m0 * 128);
    unsigned int ldsoff = (unsigned int)(unsigned long long)(void*)lds_a;

    uint4v g0;
    g0[0] = 1u;                                   // count=1 (valid user D#)
    g0[1] = ldsoff;                               // lds_addr (bytes)
    g0[2] = (unsigned int)ga;                     // global_addr[31:0]
    g0[3] = (unsigned int)((ga >> 32) & 0x01FFFFFFull) // global_addr[56:32]
            | 0x80000000u;                        // type=2 ("image")

    int8v g1;
    g1[0] = (int)(2u << 16);                      // data_size=4B; wg_mask=0
    g1[1] = (int)(128u << 16);                    // tensor_dim0 = 128 (lo16)
    g1[2] = (int)((unsigned)rows << 16);          // td0.hi=0 | tensor_dim1.lo=rows
    g1[3] = (int)(128u << 16);                    // td1.hi=0 | tile_dim0=128
    g1[4] = (int)(unsigned)rows;                  // tile_dim1=rows, tile_dim2=0
    g1[5] = 128;                                  // tensor_dim0_stride = 128
    g1[6] = 0;
    g1[7] = 0;

    int4v gz4 = {};                               // groups 2/3 unused (2D)
    int8v gz8 = {};                               // trailing group (6-arg form)
    __builtin_amdgcn_tensor_load_to_lds(g0, g1, gz4, gz4, gz8, 0);
    __builtin_amdgcn_s_wait_tensorcnt(0);
  }
  __syncthreads();

  // --- WMMA main loop -----------------------------------------------------
  const float* ap = lds_a + l15 * 128 + koff;                 // LDS A frags
  const float* bp = W + (size_t)koff * 128 + n0 + l15;        // global B frags

  v8f acc = {};
#pragma unroll
  for (int k0 = 0; k0 < 128; k0 += 4) {
    v2f a, b;
    a.x = ap[k0 + 0];
    a.y = ap[k0 + 1];
    b.x = bp[(size_t)k0 * 128];
    b.y = bp[(size_t)(k0 + 1) * 128];
    acc = __builtin_amdgcn_wmma_f32_16x16x4_f32(false, a, false, b,
                                                (short)0, acc, false, false);
  }

  // C/D layout: lanes 0-15 -> rows m0+0..7, lanes 16-31 -> rows m0+8..15
  const int rbase = m0 + (hi << 3);
  const int col = n0 + l15;
#pragma unroll
  for (int r = 0; r < 8; ++r) {
    int row = rbase + r;
    if (row < n) h[(size_t)row * 128 + col] = acc[r];
  }
}

// ---------------------------------------------------------------------------
// attention coefficients layer 1: a_src/a_dst [N,8]
// ---------------------------------------------------------------------------
__global__ __launch_bounds__(256) void k_att1(const float* __restrict__ h1,
                                              const float* __restrict__ att_s,
                                              const float* __restrict__ att_d,
                                              float* __restrict__ a_s,
                                              float* __restrict__ a_d, int N) {
  int i = blockIdx.x * blockDim.x + threadIdx.x;
  if (i >= N * 8) return;
  int node = i >> 3, hd = i & 7;
  const float* hp = h1 + (size_t)node * 128 + hd * 16;
  const float* as = att_s + hd * 16;
  const float* ad = att_d + hd * 16;
  float ss = 0.0f, sd = 0.0f;
#pragma unroll
  for (int c = 0; c < 16; ++c) { ss += hp[c] * as[c]; sd += hp[c] * ad[c]; }
  a_s[i] = ss;
  a_d[i] = sd;
}

// ---------------------------------------------------------------------------
// helpers
// ---------------------------------------------------------------------------
__device__ __forceinline__ void edge_sd(const int* __restrict__ ei, int e, int E,
                                        int& s, int& d) {
  if (e < E) { s = ei[e]; d = ei[E + e]; }
  else       { s = d = e - E; }              // self loops appended
}

__device__ __forceinline__ void atomicMaxF(float* addr, float v) {
  if (v >= 0.0f) atomicMax((int*)addr, __float_as_int(v));
  else           atomicMin((unsigned int*)addr, __float_as_uint(v));
}

__device__ __forceinline__ float lrelu(float v) {
  return v > 0.0f ? v : NEG_SLOPE * v;
}

// ---------------------------------------------------------------------------
// layer-1 edge passes (one thread per edge*head)
// ---------------------------------------------------------------------------
__global__ __launch_bounds__(256) void k_emax1(const int* __restrict__ ei, int E,
                                               int Et, const float* __restrict__ a_s,
                                               const float* __restrict__ a_d,
                                               float* __restrict__ m1) {
  int i = blockIdx.x * blockDim.x + threadIdx.x;
  if (i >= Et * 8) return;
  int e = i >> 3, hd = i & 7, s, d;
  edge_sd(ei, e, E, s, d);
  float ev = lrelu(a_s[s * 8 + hd] + a_d[d * 8 + hd]);
  atomicMaxF(&m1[d * 8 + hd], ev);
}

__global__ __launch_bounds__(256) void k_esum1(const int* __restrict__ ei, int E,
                                               int Et, const float* __restrict__ a_s,
                                               const float* __restrict__ a_d,
                                               const float* __restrict__ m1,
                                               float* __restrict__ s1) {
  int i = blockIdx.x * blockDim.x + threadIdx.x;
  if (i >= Et * 8) return;
  int e = i >> 3, hd = i & 7, s, d;
  edge_sd(ei, e, E, s, d);
  float ev = lrelu(a_s[s * 8 + hd] + a_d[d * 8 + hd]);
  atomicAdd(&s1[d * 8 + hd], expf(ev - m1[d * 8 + hd]));
}

__global__ __launch_bounds__(256) void k_eagg1(const int* __restrict__ ei, int E,
                                               int Et, const float* __restrict__ a_s,
                                               const float* __restrict__ a_d,
                                               const float* __restrict__ m1,
                                               const float* __restrict__ s1,
                                               const float* __restrict__ h1,
                                               float* __restrict__ out1) {
  int i = blockIdx.x * blockDim.x + threadIdx.x;
  if (i >= Et * 8) return;
  int e = i >> 3, hd = i & 7, s, d;
  edge_sd(ei, e, E, s, d);
  float ev = lrelu(a_s[s * 8 + hd] + a_d[d * 8 + hd]);
  float alpha = expf(ev - m1[d * 8 + hd]) / s1[d * 8 + hd];
  const float* hs = h1 + (size_t)s * 128 + hd * 16;
  float* od = out1 + (size_t)d * 128 + hd * 16;
#pragma unroll
  for (int c = 0; c < 16; ++c) atomicAdd(&od[c], hs[c] * alpha);
}

// finalize layer1: h1p = elu(out1 + b1)  (h1p overwrites h1 buffer)
__global__ __launch_bounds__(256) void k_fin1(const float* __restrict__ out1,
                                              const float* __restrict__ b1,
                                              float* __restrict__ h1p, int N) {
  int i = blockIdx.x * blockDim.x + threadIdx.x;
  if (i >= N * 128) return;
  float v = out1[i] + b1[i & 127];
  h1p[i] = v > 0.0f ? v : (expf(v) - 1.0f);
}

// ---------------------------------------------------------------------------
// layer-2 features: h2[N,2] = h1p @ W2; a2s/a2d [N]
// ---------------------------------------------------------------------------
__global__ __launch_bounds__(256) void k_feat2(const float* __restrict__ h1p,
                                               const float* __restrict__ W2,
                                               const float* __restrict__ att_s2,
                                               const float* __restrict__ att_d2,
                                               float* __restrict__ h2,
                                               float* __restrict__ a2s,
                                               float* __restrict__ a2d, int N) {
  int nidx = blockIdx.x * blockDim.x + threadIdx.x;
  if (nidx >= N) return;
  const float* hp = h1p + (size_t)nidx * 128;
  float c0 = 0.0f, c1 = 0.0f;
#pragma unroll 8
  for (int j = 0; j < 128; ++j) {
    float hv = hp[j];
    c0 += hv * W2[2 * j];
    c1 += hv * W2[2 * j + 1];
  }
  h2[2 * nidx] = c0;
  h2[2 * nidx + 1] = c1;
  a2s[nidx] = c0 * att_s2[0] + c1 * att_s2[1];
  a2d[nidx] = c0 * att_d2[0] + c1 * att_d2[1];
}

// layer-2 edge passes (one thread per edge, single head)
__global__ __launch_bounds__(256) void k_emax2(const int* __restrict__ ei, int E,
                                               int Et, const float* __restrict__ a2s,
                                               const float* __restrict__ a2d,
                                               float* __restrict__ m2) {
  int e = blockIdx.x * blockDim.x + threadIdx.x;
  if (e >= Et) return;
  int s, d;
  edge_sd(ei, e, E, s, d);
  atomicMaxF(&m2[d], lrelu(a2s[s] + a2d[d]));
}

__global__ __launch_bounds__(256) void k_esum2(const int* __restrict__ ei, int E,
                                               int Et, const float* __restrict__ a2s,
                                               const float* __restrict__ a2d,
                                               const float* __restrict__ m2,
                                               float* __restrict__ s2) {
  int e = blockIdx.x * blockDim.x + threadIdx.x;
  if (e >= Et) return;
  int s, d;
  edge_sd(ei, e, E, s, d);
  float ev = lrelu(a2s[s] + a2d[d]);
  atomicAdd(&s2[d], expf(ev - m2[d]));
}

__global__ __launch_bounds__(256) void k_eagg2(const int* __restrict__ ei, int E,
                                               int Et, const float* __restrict__ a2s,
                                               const float* __restrict__ a2d,
                                               const float* __restrict__ m2,
                                               const float* __restrict__ s2,
                                               const float* __restrict__ h2,
                                               float* __restrict__ dout) {
  int e = blockIdx.x * blockDim.x + threadIdx.x;
  if (e >= Et) return;
  int s, d;
  edge_sd(ei, e, E, s, d);
  float ev = lrelu(a2s[s] + a2d[d]);
  float alpha = expf(ev - m2[d]) / s2[d];
  atomicAdd(&dout[2 * d], h2[2 * s] * alpha);
  atomicAdd(&dout[2 * d + 1], h2[2 * s + 1] * alpha);
}

__global__ __launch_bounds__(256) void k_fin2(float* __restrict__ dout,
                                              const float* __restrict__ b2, int N) {
  int i = blockIdx.x * blockDim.x + threadIdx.x;
  if (i >= N * 2) return;
  dout[i] += b2[i & 1];
}

// ---------------------------------------------------------------------------
// launch
// ---------------------------------------------------------------------------
extern "C" void kernel_launch(void* const* d_in, const int* in_sizes, int n_in,
                              void* d_out, int out_size, void* d_ws, size_t ws_size,
                              hipStream_t stream) {
  const float* x    = (const float*)d_in[0];
  const int*   ei   = (const int*)d_in[1];
  const float* W1   = (const float*)d_in[2];
  const float* as1  = (const float*)d_in[3];
  const float* ad1  = (const float*)d_in[4];
  const float* b1   = (const float*)d_in[5];
  const float* W2   = (const float*)d_in[6];
  const float* as2  = (const float*)d_in[7];
  const float* ad2  = (const float*)d_in[8];
  const float* b2   = (const float*)d_in[9];
  float* dout = (float*)d_out;

  const int N  = in_sizes[0] / 128;
  const int E  = in_sizes[1] / 2;
  const int Et = E + N;

  float* ws  = (float*)d_ws;
  float* h1  = ws;  ws += (size_t)N * 128;   // also reused as h1p after layer1
  float* out1= ws;  ws += (size_t)N * 128;
  float* aS1 = ws;  ws += (size_t)N * 8;
  float* aD1 = ws;  ws += (size_t)N * 8;
  float* m1  = ws;  ws += (size_t)N * 8;
  float* s1  = ws;  ws += (size_t)N * 8;
  float* h2  = ws;  ws += (size_t)N * 2;
  float* a2s = ws;  ws += (size_t)N;
  float* a2d = ws;  ws += (size_t)N;
  float* m2  = ws;  ws += (size_t)N;
  float* s2  = ws;  ws += (size_t)N;

  const int B = 256;
  auto cdiv = [](long long a, long long b) { return (int)((a + b - 1) / b); };

  k_init <<<cdiv((long long)N * 128, B), B, 0, stream>>>(out1, s1, m1, s2, m2, dout, N);
  k_gemm1<<<cdiv(N, 16), 256, 0, stream>>>(x, W1, h1, N);
  k_att1 <<<cdiv((long long)N * 8, B), B, 0, stream>>>(h1, as1, ad1, aS1, aD1, N);
  k_emax1<<<cdiv((long long)Et * 8, B), B, 0, stream>>>(ei, E, Et, aS1, aD1, m1);
  k_esum1<<<cdiv((long long)Et * 8, B), B, 0, stream>>>(ei, E, Et, aS1, aD1, m1, s1);
  k_eagg1<<<cdiv((long long)Et * 8, B), B, 0, stream>>>(ei, E, Et, aS1, aD1, m1, s1, h1, out1);
  k_fin1 <<<cdiv((long long)N * 128, B), B, 0, stream>>>(out1, b1, h1, N);
  k_feat2<<<cdiv(N, B), B, 0, stream>>>(h1, W2, as2, ad2, h2, a2s, a2d, N);
  k_emax2<<<cdiv(Et, B), B, 0, stream>>>(ei, E, Et, a2s, a2d, m2);
  k_esum2<<<cdiv(Et, B), B, 0, stream>>>(ei, E, Et, a2s, a2d, m2, s2);
  k_eagg2<<<cdiv(Et, B), B, 0, stream>>>(ei, E, Et, a2s, a2d, m2, s2, h2, dout);
  k_fin2 <<<cdiv((long long)N * 2, B), B, 0, stream>>>(dout, b2, N);
}